// ConcatMesPassing_11304353923434
// MI455X (gfx1250) — compile-verified
//
#include <hip/hip_runtime.h>
#include <hip/hip_bf16.h>
#include <math.h>

// Problem constants (from reference)
#define N_NODES 4096
#define N_EDGES 16384
#define N_DIMC  128   // node feature dim
#define E_DIMC  64    // edge feature dim
#define C_DIMC  128   // context dim
#define IN_DIMC 320   // 128 + 64 + 128

typedef float v2f __attribute__((ext_vector_type(2)));
typedef float v8f __attribute__((ext_vector_type(8)));

// One K-step of the f32 WMMA GEMM: A fragment (16x4) from a contiguous float
// pair per lane, B fragments (4x16) gathered from W, NT column tiles.
template <int NCOL, int NT>
__device__ __forceinline__ void wmma_step(const float* aptr,
                                          const float* __restrict__ W,
                                          int kbase, int mrow, v8f (&acc)[NT]) {
    // A: lane (mrow, khalf) holds A[mrow][kbase], A[mrow][kbase+1]
    v2f aFrag = *(const v2f*)aptr;
#pragma unroll
    for (int t = 0; t < NT; ++t) {
        const int col = t * 16 + mrow;
        v2f bFrag;
        bFrag.x = W[(size_t)kbase * NCOL + col];
        bFrag.y = W[(size_t)(kbase + 1) * NCOL + col];
        acc[t] = __builtin_amdgcn_wmma_f32_16x16x4_f32(
            /*neg_a=*/false, aFrag, /*neg_b=*/false, bFrag,
            /*c_mod=*/(short)0, acc[t], /*reuse_a=*/false, /*reuse_b=*/false);
    }
}

// out[E, NCOL] = leaky_relu( concat(u, mid, v) @ W + bias )
// One wave per 16-row block; NCOL/16 accumulator tiles per wave; f32 WMMA.
// K loop statically split at the concat boundaries (128 / 192) so the inner
// loops have branch-free, immediate-offset addressing.
template <int NCOL>
__global__ void __launch_bounds__(256)
mlp_wmma_kernel(const float* __restrict__ u, const float* __restrict__ mid,
                const float* __restrict__ v, const float* __restrict__ W,
                const float* __restrict__ bias, float* __restrict__ out) {
    constexpr int NT = NCOL / 16;
    const int lane  = threadIdx.x & 31;
    const int wave  = threadIdx.x >> 5;
    const int rblk  = blockIdx.x * (blockDim.x >> 5) + wave;  // 16-row block id
    const int r0    = rblk * 16;
    const int mrow  = lane & 15;       // A: M index / B,C,D: N index
    const int khalf = lane >> 4;       // which K-pair half this lane holds

    v8f acc[NT];
#pragma unroll
    for (int t = 0; t < NT; ++t) acc[t] = 0.0f;

    const int e = r0 + mrow;           // A-matrix row this lane supplies
    const int ko = 2 * khalf;          // lane's K offset within a K4 step
    const float* aU = u   + (size_t)e * 128 + ko;
    const float* aM = mid + (size_t)e * 64  + ko;
    const float* aV = v   + (size_t)e * 128 + ko;

    // Segment 1: K = [0, 128)  <- u_features
#pragma unroll 4
    for (int k0 = 0; k0 < 128; k0 += 4)
        wmma_step<NCOL, NT>(aU + k0, W, k0 + ko, mrow, acc);
    // Segment 2: K = [128, 192) <- mid (edge / new_edge features)
#pragma unroll 4
    for (int k0 = 0; k0 < 64; k0 += 4)
        wmma_step<NCOL, NT>(aM + k0, W, 128 + k0 + ko, mrow, acc);
    // Segment 3: K = [192, 320) <- v_features
#pragma unroll 4
    for (int k0 = 0; k0 < 128; k0 += 4)
        wmma_step<NCOL, NT>(aV + k0, W, 192 + k0 + ko, mrow, acc);

    // Epilogue: C/D layout — VGPR i holds row m = i + 8*khalf, col n = mrow.
#pragma unroll
    for (int t = 0; t < NT; ++t) {
        const int col = t * 16 + mrow;
        const float bb = bias[col];
#pragma unroll
        for (int i = 0; i < 8; ++i) {
            const int m = i + 8 * khalf;
            float x = acc[t][i] + bb;
            x = (x > 0.0f) ? x : 0.01f * x;
            out[(size_t)(r0 + m) * NCOL + col] = x;
        }
    }
}

// a[e] = leaky_relu(new_edge[e,:] . W_a + b_a)
__global__ void __launch_bounds__(256)
attn_scalar_kernel(const float* __restrict__ ne, const float* __restrict__ Wa,
                   const float* __restrict__ ba, float* __restrict__ a) {
    const int e = blockIdx.x * blockDim.x + threadIdx.x;
    if (e >= N_EDGES) return;
    float s = ba[0];
#pragma unroll 8
    for (int k = 0; k < E_DIMC; ++k)
        s = fmaf(ne[(size_t)e * E_DIMC + k], Wa[k], s);
    a[e] = (s > 0.0f) ? s : 0.01f * s;
}

// Fused: weight = matrix>0 ? a[j] : -1e9; softmax over row; elu(weight @ neighbor).
// One 256-thread workgroup per node. Exploits exp(-1e9 - max) == 0.0f in f32,
// so only compacted live edges contribute — exactly matching the dense reference.
#define CAP 1024
__global__ void __launch_bounds__(256)
aggregate_kernel(const float* __restrict__ mat, const float* __restrict__ a,
                 const float* __restrict__ nb, float* __restrict__ out) {
    __shared__ int   s_idx[CAP];
    __shared__ float s_a[CAP];
    __shared__ int   s_cnt;
    __shared__ float s_max, s_den;

    const int node = blockIdx.x;
    if (threadIdx.x == 0) s_cnt = 0;
    __syncthreads();

    // Pass 1: stream the 64KB row with float4 loads, compact live columns.
    const float4* row4 = (const float4*)(mat + (size_t)node * N_EDGES);
    for (int q = threadIdx.x; q < N_EDGES / 4; q += blockDim.x) {
        __builtin_prefetch(row4 + q + 2 * blockDim.x, 0, 1);  // global_prefetch_b8
        float4 m4 = row4[q];
        const int j0 = q * 4;
        if (m4.x > 0.0f) { int p = atomicAdd(&s_cnt, 1); if (p < CAP) { s_idx[p] = j0 + 0; s_a[p] = a[j0 + 0]; } }
        if (m4.y > 0.0f) { int p = atomicAdd(&s_cnt, 1); if (p < CAP) { s_idx[p] = j0 + 1; s_a[p] = a[j0 + 1]; } }
        if (m4.z > 0.0f) { int p = atomicAdd(&s_cnt, 1); if (p < CAP) { s_idx[p] = j0 + 2; s_a[p] = a[j0 + 2]; } }
        if (m4.w > 0.0f) { int p = atomicAdd(&s_cnt, 1); if (p < CAP) { s_idx[p] = j0 + 3; s_a[p] = a[j0 + 3]; } }
    }
    __syncthreads();
    const int K = (s_cnt < CAP) ? s_cnt : CAP;

    // Softmax statistics over compacted set (K ~ 16; serial is cheapest).
    if (threadIdx.x == 0) {
        float m = -1e9f, den = 0.0f;
        for (int p = 0; p < K; ++p) m = fmaxf(m, s_a[p]);
        for (int p = 0; p < K; ++p) den += expf(s_a[p] - m);
        s_max = m; s_den = den;
    }
    __syncthreads();
    for (int p = threadIdx.x; p < K; p += blockDim.x)
        s_a[p] = expf(s_a[p] - s_max) / s_den;   // normalized weights
    __syncthreads();

    // Pass 2: 128 lanes each own one context channel; gather ~K neighbor rows.
    if (threadIdx.x < C_DIMC) {
        const int c = threadIdx.x;
        float acc = 0.0f;
        if (K > 0) {
            for (int p = 0; p < K; ++p)
                acc = fmaf(s_a[p], nb[(size_t)s_idx[p] * C_DIMC + c], acc);
        } else {
            // Empty row: all weights equal exp(0)/E -> mean of neighbor features.
            for (int j = 0; j < N_EDGES; ++j) acc += nb[(size_t)j * C_DIMC + c];
            acc *= (1.0f / (float)N_EDGES);
        }
        acc = (acc > 0.0f) ? acc : (expf(acc) - 1.0f);   // elu
        out[(size_t)node * C_DIMC + c] = acc;
    }
}

extern "C" void kernel_launch(void* const* d_in, const int* in_sizes, int n_in,
                              void* d_out, int out_size, void* d_ws, size_t ws_size,
                              hipStream_t stream) {
    (void)in_sizes; (void)n_in; (void)out_size; (void)ws_size;
    const float* u   = (const float*)d_in[0];   // [E,128]
    const float* v   = (const float*)d_in[1];   // [E,128]
    const float* ef  = (const float*)d_in[2];   // [E,64]
    const float* mat = (const float*)d_in[3];   // [N,E]
    // d_in[4] (node_edge_mask) intentionally unread: derivable from mat>0.
    const float* We  = (const float*)d_in[5];   // [320,64]
    const float* be  = (const float*)d_in[6];   // [64]
    const float* W   = (const float*)d_in[7];   // [320,128]
    const float* b   = (const float*)d_in[8];   // [128]
    const float* Wa  = (const float*)d_in[9];   // [64,1]
    const float* ba  = (const float*)d_in[10];  // [1]

    float* out      = (float*)d_out;
    float* context  = out;                                   // [N,128]
    float* new_edge = out + (size_t)N_NODES * C_DIMC;        // [E,64] (2nd output)

    float* ws       = (float*)d_ws;
    float* a_vec    = ws;                                    // [E]
    float* neighbor = ws + N_EDGES;                          // [E,128]

    // 1024 16-row blocks, 8 waves/block -> 128 workgroups of 256.
    const int rowBlocks = N_EDGES / 16;
    const int wgs = rowBlocks / 8;

    mlp_wmma_kernel<E_DIMC><<<wgs, 256, 0, stream>>>(u, ef, v, We, be, new_edge);
    attn_scalar_kernel<<<N_EDGES / 256, 256, 0, stream>>>(new_edge, Wa, ba, a_vec);
    mlp_wmma_kernel<C_DIMC><<<wgs, 256, 0, stream>>>(u, new_edge, v, W, b, neighbor);
    aggregate_kernel<<<N_NODES, 256, 0, stream>>>(mat, a_vec, neighbor, context);
}